// FracDMSelfAttention_33509334843618
// MI455X (gfx1250) — compile-verified
//
#include <hip/hip_runtime.h>
#include <math.h>
#include <stdint.h>

// Problem constants (B=2, S=1024, E=256, NH=16, HD=16, BN=2048)
#define SS   1024
#define EE   256
#define NHH  16
#define HDD  16
#define BNN  2048
#define MT   (BNN / 16)      // 128 m-tiles per sweep

typedef __attribute__((ext_vector_type(2))) float v2f;
typedef __attribute__((ext_vector_type(8))) float v8f;
typedef __attribute__((ext_vector_type(4))) uint32_t u32x4;
typedef __attribute__((ext_vector_type(8))) uint32_t u32x8;

// f32 WMMA, D(16x16) = A(16x4) * B(4x16) + C.
__device__ __forceinline__ v8f wmma_f32_k4(v2f a, v2f b, v8f c) {
  return __builtin_amdgcn_wmma_f32_16x16x4_f32(false, a, false, b, (short)0, c,
                                               false, false);
}

// Fragment conventions (wave32): half = lane>>4, l15 = lane&15
//  A (16x4):  M = l15; vgpr0 = A[M][2*half], vgpr1 = A[M][2*half+1]
//  B (4x16):  N = l15; vgpr0 = B[2*half][N], vgpr1 = B[2*half+1][N]
//  C/D:       N = l15; vgpr r = C[r + 8*half][N]

// ds_swizzle (group-of-32): offset = and | (or<<5) | (xor<<10)
__device__ __forceinline__ float half_reduce_add(float x) {
  // xor 1/2/4/8 stays within each 16-lane half
  x += __int_as_float(__builtin_amdgcn_ds_swizzle(__float_as_int(x), 0x041F));
  x += __int_as_float(__builtin_amdgcn_ds_swizzle(__float_as_int(x), 0x081F));
  x += __int_as_float(__builtin_amdgcn_ds_swizzle(__float_as_int(x), 0x101F));
  x += __int_as_float(__builtin_amdgcn_ds_swizzle(__float_as_int(x), 0x201F));
  return x;
}
__device__ __forceinline__ float bcast_half(float x) {
  // every lane reads lane (lane & 16): broadcast l15==0 value within each half
  return __int_as_float(__builtin_amdgcn_ds_swizzle(__float_as_int(x), 0x0010));
}

// ---- CDNA5 async-tensor data movement (inline asm; ASYNCcnt / TENSORcnt) ----
__device__ __forceinline__ void async_ld_b128(uint32_t lds_off, uint32_t goff,
                                              uint64_t sbase) {
  asm volatile("global_load_async_to_lds_b128 %0, %1, %2"
               :: "v"(lds_off), "v"(goff), "s"(sbase) : "memory");
}
__device__ __forceinline__ void wait_async2() {
  asm volatile("s_wait_asynccnt 0x2" ::: "memory");
}
__device__ __forceinline__ void wait_async0() {
  asm volatile("s_wait_asynccnt 0x0" ::: "memory");
}
__device__ __forceinline__ void wait_ds0() {
  asm volatile("s_wait_dscnt 0x0" ::: "memory");
}

// TDM: load a 16x16 f32 tile (row stride = stride_elems) into LDS.
// D# packing per CDNA5 ISA 8.3/8.4: group0 = {count=1, lds_addr,
// global_addr[56:0], type=2}; group1 = {data_size=4B, dims=16x16,
// tile=16x16, dim0_stride}.
__device__ __forceinline__ void tdm_load_16x16_f32(uint32_t lds_addr,
                                                   const float* gptr,
                                                   uint32_t stride_elems) {
  uint64_t ga = (uint64_t)(uintptr_t)gptr;
  u32x4 g0;
  g0[0] = 1u;                                             // count=1
  g0[1] = lds_addr;                                       // lds_addr (bytes)
  g0[2] = (uint32_t)ga;                                   // global_addr[31:0]
  g0[3] = ((uint32_t)(ga >> 32) & 0x01FFFFFFu) | (2u << 30); // [56:32] | type=2
  u32x8 g1;
  g1[0] = (2u << 16);        // wg_mask=0, data_size=2 (4 bytes)
  g1[1] = (16u << 16);       // tensor_dim0 = 16 (bits 79:48, low half)
  g1[2] = (16u << 16);       // tensor_dim1 = 16 (bits 111:80, low half)
  g1[3] = (16u << 16);       // tile_dim0 = 16 (bits 127:112)
  g1[4] = 16u;               // tile_dim1 = 16, tile_dim2 = 0
  g1[5] = stride_elems;      // tensor_dim0_stride[31:0]
  g1[6] = 0u;                // stride hi / dim1_stride lo
  g1[7] = 0u;
  asm volatile("tensor_load_to_lds %0, %1" :: "s"(g0), "s"(g1) : "memory");
}

// ---------------------------------------------------------------------------
// Kernel 1: q/v projections via WMMA + per-row L2 normalize of q, scattered
// into head-remixed layout: r = b*16384+s*16+hh; h=r>>11; n=r&2047; Qh[h][n][d]
// V natural: V[m = b*1024+s][hh][d]
// ---------------------------------------------------------------------------
__global__ void __launch_bounds__(32)
proj_norm_kernel(const float* __restrict__ X,
                 const float* __restrict__ Wq, const float* __restrict__ bq,
                 const float* __restrict__ Wv, const float* __restrict__ bv,
                 float* __restrict__ Qh, float* __restrict__ Vb)
{
  const int qtiles = (BNN / 16) * (EE / 16);   // 2048
  const int tile   = blockIdx.x;
  const bool is_q  = tile < qtiles;
  const int t2     = is_q ? tile : tile - qtiles;
  const int rowTile = t2 >> 4;
  const int colTile = t2 & 15;
  const int row0 = rowTile * 16;
  const int col0 = colTile * 16;
  const int lane = threadIdx.x & 31;
  const int half = lane >> 4;
  const int l15  = lane & 15;
  const float* __restrict__ W    = is_q ? Wq : Wv;
  const float* __restrict__ bias = is_q ? bq : bv;

  v8f acc = {};
  for (int k0 = 0; k0 < EE; k0 += 4) {
    const int k = k0 + 2 * half;
    v2f a, b;
    a.x = X[(row0 + l15) * EE + k];
    a.y = X[(row0 + l15) * EE + k + 1];
    b.x = W[(col0 + l15) * EE + k];          // B[k][n] = W[n][k]
    b.y = W[(col0 + l15) * EE + k + 1];
    acc = wmma_f32_k4(a, b, acc);
  }

  const float bN = bias[col0 + l15];
  float vals[8];
#pragma unroll
  for (int r = 0; r < 8; ++r) vals[r] = acc[r] + bN;

  if (is_q) {
#pragma unroll
    for (int r = 0; r < 8; ++r) {
      const float s2 = half_reduce_add(vals[r] * vals[r]);
      vals[r] *= 1.0f / fmaxf(sqrtf(s2), 1e-12f);
    }
#pragma unroll
    for (int r = 0; r < 8; ++r) {
      const int token = row0 + r + 8 * half;                 // b*1024 + s
      const int ridx  = (token >> 10) * 16384 + (token & 1023) * 16 + colTile;
      const int h = ridx >> 11;
      const int n = ridx & 2047;
      Qh[(h * BNN + n) * HDD + l15] = vals[r];
    }
  } else {
#pragma unroll
    for (int r = 0; r < 8; ++r) {
      const int token = row0 + r + 8 * half;
      Vb[token * EE + col0 + l15] = vals[r];
    }
  }
}

// ---------------------------------------------------------------------------
// Kernel 2 (pass 1): inv[h][n] = 1 / ( C * sum_m exp(sqrt(max(2-2*dot,0))) )
// 4 waves/block share a double-buffered Q m-tile staged with async-to-LDS.
// Row sums via WMMA against a ones-column B fragment.
// ---------------------------------------------------------------------------
__global__ void __launch_bounds__(128)
pass1_kernel(const float* __restrict__ Qh, float* __restrict__ invq)
{
  __shared__ __attribute__((aligned(16))) float qbuf[2][256];
  __shared__ __attribute__((aligned(16))) float wst[4][256];

  const int blk  = blockIdx.x;                 // 0..511
  const int h    = blk >> 5;                   // 32 blocks / head
  const int wave = threadIdx.x >> 5;
  const int lane = threadIdx.x & 31;
  const int half = lane >> 4;
  const int l15  = lane & 15;
  const int n0   = ((blk & 31) * 4 + wave) * 16;
  const float* __restrict__ Q = Qh + (size_t)h * BNN * HDD;
  const uint64_t Qg = (uint64_t)(uintptr_t)Q;
  float* __restrict__ wme = &wst[wave][0];

  v2f aq[4];
#pragma unroll
  for (int ks = 0; ks < 4; ++ks) {
    const int k = ks * 4 + 2 * half;
    aq[ks].x = Q[(n0 + l15) * HDD + k];
    aq[ks].y = Q[(n0 + l15) * HDD + k + 1];
  }
  v2f bones;
  bones.x = (l15 == 0) ? 1.0f : 0.0f;
  bones.y = bones.x;

  const uint32_t qlds[2] = {(uint32_t)(uintptr_t)&qbuf[0][0],
                            (uint32_t)(uintptr_t)&qbuf[1][0]};
  if (wave == 0) {   // prologue: m-tile 0 (1 KB contiguous) -> qbuf[0]
    async_ld_b128(qlds[0] + lane * 16, (uint32_t)(lane * 16), Qg);
    async_ld_b128(qlds[0] + 512 + lane * 16, (uint32_t)(512 + lane * 16), Qg);
  }

  v8f rs = {};
  for (int it = 0; it < MT; ++it) {
    const int cur = it & 1;
    if (wave == 0) {
      if (it + 1 < MT) {
        const uint32_t go = (uint32_t)((it + 1) * 16 * HDD * 4);
        async_ld_b128(qlds[cur ^ 1] + lane * 16, go + lane * 16, Qg);
        async_ld_b128(qlds[cur ^ 1] + 512 + lane * 16, go + 512 + lane * 16, Qg);
        wait_async2();                    // current tile's 2 loads complete
      } else {
        wait_async0();
      }
    }
    __syncthreads();                      // publish qbuf[cur]

    const float* __restrict__ qc = &qbuf[cur][0];
    v8f dc = {};
#pragma unroll
    for (int ks = 0; ks < 4; ++ks) {
      const int k = ks * 4 + 2 * half;
      v2f b;                              // B[k][m] = Qtile[m][k]
      b.x = qc[l15 * HDD + k];
      b.y = qc[l15 * HDD + k + 1];
      dc = wmma_f32_k4(aq[ks], b, dc);
    }
#pragma unroll
    for (int r = 0; r < 8; ++r) {
      const float d2 = fmaxf(2.0f - 2.0f * dc[r], 0.0f);
      wme[(r + 8 * half) * 16 + l15] = __expf(sqrtf(d2));
    }
    wait_ds0();                           // wave-local staging ready
#pragma unroll
    for (int ks = 0; ks < 4; ++ks) {
      const int k = ks * 4 + 2 * half;
      v2f a;
      a.x = wme[l15 * 16 + k];
      a.y = wme[l15 * 16 + k + 1];
      rs = wmma_f32_k4(a, bones, rs);
    }
    __syncthreads();                      // all reads of qbuf[cur] done
  }

  if (l15 == 0) {
    const float Cc = 2.118516e-14f;       // (2*pi*pi)^(-8) / 2048
#pragma unroll
    for (int r = 0; r < 8; ++r) {
      const int n = n0 + r + 8 * half;
      invq[h * BNN + n] = 1.0f / (Cc * rs[r]);
    }
  }
}

// ---------------------------------------------------------------------------
// Kernel 3 (pass 2): out[h][n][:] = (sum_m K*inv[m]*v[m,h,:]) / (sum_m K*inv[m])
// K = (1+acos(dot)/sqrt(pi))^-17 via 4 squarings + v_rcp. Q m-tile via
// async-to-LDS (wave0), strided V m-tile via TDM tensor_load_to_lds (wave1).
// ---------------------------------------------------------------------------
__global__ void __launch_bounds__(128)
pass2_kernel(const float* __restrict__ Qh, const float* __restrict__ Vb,
             const float* __restrict__ invq, float* __restrict__ out)
{
  __shared__ __attribute__((aligned(16))) float qbuf[2][256];
  __shared__ __attribute__((aligned(16))) float vbuf[2][256];
  __shared__ __attribute__((aligned(16))) float wst[4][256];

  const int blk  = blockIdx.x;
  const int h    = blk >> 5;
  const int wave = threadIdx.x >> 5;
  const int lane = threadIdx.x & 31;
  const int half = lane >> 4;
  const int l15  = lane & 15;
  const int n0   = ((blk & 31) * 4 + wave) * 16;
  const float* __restrict__ Q  = Qh + (size_t)h * BNN * HDD;
  const float* __restrict__ Vh = Vb + h * HDD;      // + m*EE + d
  const uint64_t Qg = (uint64_t)(uintptr_t)Q;
  float* __restrict__ wme = &wst[wave][0];

  v2f aq[4];
#pragma unroll
  for (int ks = 0; ks < 4; ++ks) {
    const int k = ks * 4 + 2 * half;
    aq[ks].x = Q[(n0 + l15) * HDD + k];
    aq[ks].y = Q[(n0 + l15) * HDD + k + 1];
  }
  v2f bones;
  bones.x = (l15 == 0) ? 1.0f : 0.0f;
  bones.y = bones.x;

  const uint32_t qlds[2] = {(uint32_t)(uintptr_t)&qbuf[0][0],
                            (uint32_t)(uintptr_t)&qbuf[1][0]};
  const uint32_t vlds[2] = {(uint32_t)(uintptr_t)&vbuf[0][0],
                            (uint32_t)(uintptr_t)&vbuf[1][0]};
  if (wave == 0) {
    async_ld_b128(qlds[0] + lane * 16, (uint32_t)(lane * 16), Qg);
    async_ld_b128(qlds[0] + 512 + lane * 16, (uint32_t)(512 + lane * 16), Qg);
  } else if (wave == 1) {
    tdm_load_16x16_f32(vlds[0], Vh, EE);
  }

  v8f num = {};
  v8f den = {};
  const float invSqrtPi = 0.56418958354775628f;

  for (int it = 0; it < MT; ++it) {
    const int m0  = it * 16;
    const int cur = it & 1;
    if (wave == 0) {
      if (it + 1 < MT) {
        const uint32_t go = (uint32_t)((it + 1) * 16 * HDD * 4);
        async_ld_b128(qlds[cur ^ 1] + lane * 16, go + lane * 16, Qg);
        async_ld_b128(qlds[cur ^ 1] + 512 + lane * 16, go + 512 + lane * 16, Qg);
        wait_async2();
      } else {
        wait_async0();
      }
    } else if (wave == 1) {
      if (it + 1 < MT) {
        tdm_load_16x16_f32(vlds[cur ^ 1], Vh + (size_t)(m0 + 16) * EE, EE);
        __builtin_amdgcn_s_wait_tensorcnt((short)1);
      } else {
        __builtin_amdgcn_s_wait_tensorcnt((short)0);
      }
    }
    __syncthreads();                      // publish qbuf[cur] / vbuf[cur]

    const float* __restrict__ qc = &qbuf[cur][0];
    const float* __restrict__ vc = &vbuf[cur][0];

    v8f dc = {};
#pragma unroll
    for (int ks = 0; ks < 4; ++ks) {
      const int k = ks * 4 + 2 * half;
      v2f b;
      b.x = qc[l15 * HDD + k];
      b.y = qc[l15 * HDD + k + 1];
      dc = wmma_f32_k4(aq[ks], b, dc);
    }
    const float invm = invq[h * BNN + m0 + l15];
#pragma unroll
    for (int r = 0; r < 8; ++r) {
      const float x    = fminf(fmaxf(dc[r], -1.0f), 1.0f);
      const float ac   = acosf(x);                  // acos(1)=0, acos(-1)=pi
      const float base = fmaf(ac, invSqrtPi, 1.0f); // >= 1
      const float p2 = base * base;
      const float p4 = p2 * p2;
      const float p8 = p4 * p4;
      const float p16 = p8 * p8;
      const float Kv = __builtin_amdgcn_rcpf(p16 * base);  // base^-17
      wme[(r + 8 * half) * 16 + l15] = Kv * invm;
    }
    wait_ds0();
#pragma unroll
    for (int ks = 0; ks < 4; ++ks) {
      const int k = ks * 4 + 2 * half;
      v2f a;                                        // W as A (M=n, K=m)
      a.x = wme[l15 * 16 + k];
      a.y = wme[l15 * 16 + k + 1];
      v2f b;                                        // B[k][d] = Vtile[k][d]
      b.x = vc[k * 16 + l15];
      b.y = vc[(k + 1) * 16 + l15];
      num = wmma_f32_k4(a, b, num);
      den = wmma_f32_k4(a, bones, den);
    }
    __syncthreads();
  }

#pragma unroll
  for (int r = 0; r < 8; ++r) {
    const float dsum = bcast_half(den[r]);          // row sum sits at l15==0
    const float val  = num[r] / dsum;
    const int nrow = n0 + r + 8 * half;
    const int ridx = h * BNN + nrow;
    const int b_ = ridx >> 14;
    const int s_ = (ridx >> 4) & 1023;
    const int hh = ridx & 15;
    out[(b_ * SS + s_) * EE + hh * HDD + l15] = val;
  }
}

// ---------------------------------------------------------------------------
extern "C" void kernel_launch(void* const* d_in, const int* in_sizes, int n_in,
                              void* d_out, int out_size, void* d_ws, size_t ws_size,
                              hipStream_t stream) {
  // setup_inputs order: hidden_states, g, attention_mask, Wq, bq, Wk, bk, Wv, bv
  const float* X  = (const float*)d_in[0];
  const float* Wq = (const float*)d_in[3];
  const float* bq = (const float*)d_in[4];
  const float* Wv = (const float*)d_in[7];
  const float* bv = (const float*)d_in[8];
  // g / attention_mask / Wk / bk are dead code in the reference output.

  float* ws   = (float*)d_ws;
  float* Qh   = ws;                       // NHH*BNN*HDD  = 524288 f32 (2 MB)
  float* Vb   = Qh + NHH * BNN * HDD;     // BNN*EE       = 524288 f32 (2 MB)
  float* invq = Vb + BNN * EE;            // NHH*BNN      =  32768 f32 (128 KB)
  float* out  = (float*)d_out;            // 2*1024*256 f32

  proj_norm_kernel<<<2 * (BNN / 16) * (EE / 16), 32, 0, stream>>>(
      X, Wq, bq, Wv, bv, Qh, Vb);
  pass1_kernel<<<NHH * (MT / 4), 128, 0, stream>>>(Qh, invq);
  pass2_kernel<<<NHH * (MT / 4), 128, 0, stream>>>(Qh, Vb, invq, out);
}